// GNN_58334245814860
// MI455X (gfx1250) — compile-verified
//
#include <hip/hip_runtime.h>

#define DIM 128

typedef float v2f __attribute__((ext_vector_type(2)));
typedef float v8f __attribute__((ext_vector_type(8)));

// ---------------------------------------------------------------------------
// deg[i] = 1.0  (self-loop weight)
__global__ void GNN_deg_init(float* __restrict__ deg, int n) {
    int i = blockIdx.x * blockDim.x + threadIdx.x;
    if (i < n) deg[i] = 1.0f;
}

// deg[dst[e]] += w[e]
__global__ void GNN_deg_accum(const int* __restrict__ dst,
                              const float* __restrict__ w,
                              float* __restrict__ deg, int e) {
    int i = blockIdx.x * blockDim.x + threadIdx.x;
    if (i < e) atomicAdd(&deg[dst[i]], w[i]);
}

// deg -> dis = deg>0 ? rsqrt(deg) : 0   (in place)
__global__ void GNN_dis(float* __restrict__ deg, int n) {
    int i = blockIdx.x * blockDim.x + threadIdx.x;
    if (i < n) {
        float d = deg[i];
        deg[i] = (d > 0.0f) ? rsqrtf(d) : 0.0f;
    }
}

// ---------------------------------------------------------------------------
// h = x @ W  via V_WMMA_F32_16X16X4_F32.
// One wave computes one 16x16 tile of h; block = 8 waves = 8 column tiles
// (DIM/16 = 8), grid.x = row tiles. W (64KB) stays cache-resident.
__global__ __launch_bounds__(256) void GNN_gemm_wmma(
        const float* __restrict__ x,     // [n, DIM] row-major
        const float* __restrict__ Wm,    // [DIM, DIM] row-major
        float* __restrict__ h,           // [n, DIM]
        int n) {
    const int lane = threadIdx.x & 31;
    const int wave = threadIdx.x >> 5;
    const int m0   = blockIdx.x * 16;        // row tile base
    const int n0   = wave * 16;              // col tile base
    const int rc   = lane & 15;              // row (for A) / col (for B & C)
    const int kb   = (lane >> 4) << 1;       // lanes 0-15: K 0,1 ; lanes 16-31: K 2,3

    // clamp load row so EXEC stays all-1s for WMMA even on a ragged tail
    int arow = m0 + rc;
    if (arow >= n) arow = n - 1;

    const float* __restrict__ xa = x  + (size_t)arow * DIM + kb;
    const float* __restrict__ wb = Wm + (size_t)kb * DIM + n0 + rc;

    v8f c = {};
#pragma unroll
    for (int k0 = 0; k0 < DIM; k0 += 4) {
        v2f a;
        a.x = xa[k0];
        a.y = xa[k0 + 1];
        v2f b;
        b.x = wb[(size_t)k0 * DIM];
        b.y = wb[(size_t)(k0 + 1) * DIM];
        // D = A(16x4,f32) * B(4x16,f32) + C
        c = __builtin_amdgcn_wmma_f32_16x16x4_f32(
                /*neg_a=*/false, a, /*neg_b=*/false, b,
                /*c_mod=*/(short)0, c, /*reuse_a=*/false, /*reuse_b=*/false);
    }

    // C/D layout: VGPR r -> M = r (lanes 0-15) or r+8 (lanes 16-31), N = rc
    const int mb = (lane >> 4) << 3;         // 0 or 8
#pragma unroll
    for (int r = 0; r < 8; ++r) {
        int m = m0 + mb + r;
        if (m < n) h[(size_t)m * DIM + n0 + rc] = c[r];
    }
}

// ---------------------------------------------------------------------------
// out[i,:] = dis[i]^2 * h[i,:] + b[:]   (self-loop message + bias)
__global__ void GNN_self_bias(const float* __restrict__ h,
                              const float* __restrict__ dis,
                              const float* __restrict__ bias,
                              float* __restrict__ out, int total) {
    int idx = blockIdx.x * blockDim.x + threadIdx.x;
    if (idx < total) {
        int node = idx >> 7;            // / DIM
        int d    = idx & (DIM - 1);
        float s  = dis[node];
        out[idx] = s * s * h[idx] + bias[d];
    }
}

// ---------------------------------------------------------------------------
// One wave32 per edge: norm = dis[src]*w*dis[dst];
// out[dst,:] += norm * h[src,:]  (lane owns a contiguous float4 of the 128 dims)
__global__ __launch_bounds__(256) void GNN_scatter(
        const int* __restrict__ src, const int* __restrict__ dst,
        const float* __restrict__ w, const float* __restrict__ dis,
        const float* __restrict__ h, float* __restrict__ out, int e) {
    int gw   = (blockIdx.x * blockDim.x + threadIdx.x) >> 5;
    int lane = threadIdx.x & 31;
    if (gw >= e) return;

    int   s    = src[gw];
    int   t    = dst[gw];
    float norm = dis[s] * w[gw] * dis[t];

    const float4 hv = ((const float4*)(h + (size_t)s * DIM))[lane];
    float* op = out + (size_t)t * DIM + lane * 4;
    atomicAdd(op + 0, norm * hv.x);
    atomicAdd(op + 1, norm * hv.y);
    atomicAdd(op + 2, norm * hv.z);
    atomicAdd(op + 3, norm * hv.w);
}

// ---------------------------------------------------------------------------
__global__ void GNN_relu(float* __restrict__ out, int total) {
    int i = blockIdx.x * blockDim.x + threadIdx.x;
    if (i < total) out[i] = fmaxf(out[i], 0.0f);
}

// ---------------------------------------------------------------------------
extern "C" void kernel_launch(void* const* d_in, const int* in_sizes, int n_in,
                              void* d_out, int out_size, void* d_ws, size_t ws_size,
                              hipStream_t stream) {
    const float* x    = (const float*)d_in[0];   // [N, DIM]
    const int*   eidx = (const int*)  d_in[1];   // [2, E]
    const float* ew   = (const float*)d_in[2];   // [E]
    const float* Wm   = (const float*)d_in[3];   // [DIM, DIM]
    const float* bias = (const float*)d_in[4];   // [DIM]
    float*       out  = (float*)d_out;           // [N, DIM]

    const int n = in_sizes[0] / DIM;             // 100000
    const int e = in_sizes[2];                   // 640000
    const int* src = eidx;
    const int* dst = eidx + e;

    // workspace: h [n*DIM floats] | deg/dis [n floats]
    float* h   = (float*)d_ws;
    float* deg = h + (size_t)n * DIM;

    const int B = 256;
    GNN_deg_init <<<(n + B - 1) / B, B, 0, stream>>>(deg, n);
    GNN_deg_accum<<<(e + B - 1) / B, B, 0, stream>>>(dst, ew, deg, e);
    GNN_dis      <<<(n + B - 1) / B, B, 0, stream>>>(deg, n);

    GNN_gemm_wmma<<<(n + 15) / 16, B, 0, stream>>>(x, Wm, h, n);

    const int total = n * DIM;
    GNN_self_bias<<<(total + B - 1) / B, B, 0, stream>>>(h, deg, bias, out, total);

    // one wave (32 threads) per edge
    long long sthreads = (long long)e * 32;
    GNN_scatter<<<(int)((sthreads + B - 1) / B), B, 0, stream>>>(
        src, dst, ew, deg, h, out, e);

    GNN_relu<<<(total + B - 1) / B, B, 0, stream>>>(out, total);
}